// GraphSAGE_51187420233989
// MI455X (gfx1250) — compile-verified
//
#include <hip/hip_runtime.h>

typedef __attribute__((ext_vector_type(2))) float v2f;
typedef __attribute__((ext_vector_type(8))) float v8f;

#define N_NODES   50000
#define N_EDGES   800000
#define NUM_GRAPHS 128
#define MTILES    (N_NODES / 16)   // 3125, exact

// ---------------- utility kernels ----------------

__global__ void zero_f32(float* __restrict__ p, int n) {
  int i = blockIdx.x * blockDim.x + threadIdx.x;
  if (i < n) p[i] = 0.0f;
}

__global__ void deg_kernel(const int* __restrict__ dst, float* __restrict__ deg) {
  int e = blockIdx.x * blockDim.x + threadIdx.x;
  if (e < N_EDGES) atomicAdd(&deg[dst[e]], 1.0f);
}

__global__ void invdeg_kernel(float* __restrict__ deg) {
  int i = blockIdx.x * blockDim.x + threadIdx.x;
  if (i < N_NODES) deg[i] = 1.0f / fmaxf(deg[i], 1.0f);
}

// Scatter-add x[src] into agg[dst]; one thread per (edge, 4-float chunk).
// x rows are 96 floats = 24 chunks; 24 consecutive threads stream one edge row.
__global__ void scatter_kernel(const float* __restrict__ x,
                               const int* __restrict__ src,
                               const int* __restrict__ dst,
                               float* __restrict__ agg) {
  const int CH = 96 / 4;
  long long idx = (long long)blockIdx.x * blockDim.x + threadIdx.x;
  if (idx >= (long long)N_EDGES * CH) return;
  int e = (int)(idx / CH);
  int c = (int)(idx % CH);
  int s = src[e];
  int d = dst[e];
  const float4 v = *(const float4*)(x + (size_t)s * 96 + c * 4);
  float* ap = agg + (size_t)d * 96 + c * 4;
  atomicAdd(ap + 0, v.x);
  atomicAdd(ap + 1, v.y);
  atomicAdd(ap + 2, v.z);
  atomicAdd(ap + 3, v.w);
}

// ---------------- WMMA GEMM ----------------
// out[m][n] = act( (agg[m]*invdeg[m]) @ Wl^T + b + xin @ Wr^T )
// One wave per (16-node mtile, 16-channel ntile). V_WMMA_F32_16X16X4_F32, K step 4.
// A 16x4 layout: lanes 0-15 M=lane, VGPR0=K even, VGPR1=K odd; lanes 16-31 take K+2 pair.
// B 4x16 mirror: lane n-column, contiguous K pair -> W[n][k..k+1] (row-major W = B^T).
__global__ void __launch_bounds__(128)
sage_gemm(const float* __restrict__ xin, const float* __restrict__ agg,
          const float* __restrict__ invdeg,
          const float* __restrict__ Wl, const float* __restrict__ bias,
          const float* __restrict__ Wr,
          float* __restrict__ out, int din, int dout, int ntiles, int do_relu) {
  int lane = threadIdx.x & 31;
  int wave = threadIdx.x >> 5;
  int tile = blockIdx.x * (blockDim.x >> 5) + wave;
  int total = MTILES * ntiles;
  if (tile >= total) return;                 // wave-uniform: EXEC stays all-1s
  int mtile = tile / ntiles;
  int ntile = tile - mtile * ntiles;

  int l15  = lane & 15;
  int kgrp = lane >> 4;                      // 0 or 1 -> K pair offset 0 / 2
  int mrow = mtile * 16 + l15;               // A row for this lane
  int ncol = ntile * 16 + l15;               // B column / C column for this lane

  float idg = invdeg[mrow];
  float bv  = bias[ncol];
  v8f c;
#pragma unroll
  for (int r = 0; r < 8; ++r) c[r] = bv;     // C init = bias broadcast over M

  const float* arow  = agg + (size_t)mrow * din;
  const float* xrow  = xin + (size_t)mrow * din;
  const float* wlrow = Wl  + (size_t)ncol * din;
  const float* wrrow = Wr  + (size_t)ncol * din;

  for (int k0 = 0; k0 < din; k0 += 4) {
    int ka = k0 + 2 * kgrp;
    v2f am = *(const v2f*)(arow + ka);
    am = am * idg;                           // fuse mean = agg * (1/deg)
    v2f ax = *(const v2f*)(xrow + ka);
    v2f bl = *(const v2f*)(wlrow + ka);
    v2f br = *(const v2f*)(wrrow + ka);
    c = __builtin_amdgcn_wmma_f32_16x16x4_f32(false, am, false, bl,
                                              (short)0, c, false, false);
    c = __builtin_amdgcn_wmma_f32_16x16x4_f32(false, ax, false, br,
                                              (short)0, c, false, false);
  }

  // C/D layout: VGPR r -> M = r + 8*kgrp, N = ncol
  float* op = out + (size_t)(mtile * 16 + 8 * kgrp) * dout + ncol;
#pragma unroll
  for (int r = 0; r < 8; ++r) {
    float v = c[r];
    if (do_relu) v = fmaxf(v, 0.0f);
    op[(size_t)r * dout] = v;
  }
}

// ---------------- pooling ----------------

__global__ void pool_kernel(const float* __restrict__ h, const int* __restrict__ batch,
                            float* __restrict__ out, float* __restrict__ cnt) {
  long long idx = (long long)blockIdx.x * blockDim.x + threadIdx.x;
  if (idx >= (long long)N_NODES * 64) return;
  int n = (int)(idx >> 6);
  int o = (int)(idx & 63);
  int g = batch[n];
  atomicAdd(&out[g * 64 + o], h[(size_t)n * 64 + o]);
  if (o == 0) atomicAdd(&cnt[g], 1.0f);
}

__global__ void pool_div(float* __restrict__ out, const float* __restrict__ cnt) {
  int idx = blockIdx.x * blockDim.x + threadIdx.x;
  if (idx < NUM_GRAPHS * 64) out[idx] /= fmaxf(cnt[idx >> 6], 1.0f);
}

// ---------------- launch ----------------

extern "C" void kernel_launch(void* const* d_in, const int* in_sizes, int n_in,
                              void* d_out, int out_size, void* d_ws, size_t ws_size,
                              hipStream_t stream) {
  (void)in_sizes; (void)n_in; (void)out_size; (void)ws_size;

  const float* x     = (const float*)d_in[0];
  const int*   edges = (const int*)d_in[1];     // [2, E]
  const int*   batch = (const int*)d_in[2];
  const float* Wl0 = (const float*)d_in[3];
  const float* b0  = (const float*)d_in[4];
  const float* Wr0 = (const float*)d_in[5];
  const float* Wl1 = (const float*)d_in[6];
  const float* b1  = (const float*)d_in[7];
  const float* Wr1 = (const float*)d_in[8];
  const float* Wl2 = (const float*)d_in[9];
  const float* b2  = (const float*)d_in[10];
  const float* Wr2 = (const float*)d_in[11];

  const int* src = edges;
  const int* dst = edges + N_EDGES;

  float* ws  = (float*)d_ws;
  float* agg = ws;                       // N*96
  float* h1  = agg + (size_t)N_NODES * 96;
  float* h2  = h1  + (size_t)N_NODES * 96;
  float* h3  = h2  + (size_t)N_NODES * 96;   // N*64
  float* deg = h3  + (size_t)N_NODES * 64;   // N (becomes invdeg)
  float* cnt = deg + (size_t)N_NODES;        // NUM_GRAPHS
  float* out = (float*)d_out;                // [128, 64]

  const int B = 256;
  const int aggN   = N_NODES * 96;
  const long long scatN = (long long)N_EDGES * 24;
  const long long poolN = (long long)N_NODES * 64;

  // graph-invariant prep + clears
  zero_f32<<<(N_NODES + B - 1) / B, B, 0, stream>>>(deg, N_NODES);
  zero_f32<<<1, 128, 0, stream>>>(cnt, NUM_GRAPHS);
  zero_f32<<<(NUM_GRAPHS * 64 + B - 1) / B, B, 0, stream>>>(out, NUM_GRAPHS * 64);
  deg_kernel<<<(N_EDGES + B - 1) / B, B, 0, stream>>>(dst, deg);
  invdeg_kernel<<<(N_NODES + B - 1) / B, B, 0, stream>>>(deg);

  // layer 0: x(96) -> h1(96), relu
  zero_f32<<<(aggN + B - 1) / B, B, 0, stream>>>(agg, aggN);
  scatter_kernel<<<(int)((scatN + B - 1) / B), B, 0, stream>>>(x, src, dst, agg);
  sage_gemm<<<(MTILES * 6 + 3) / 4, 128, 0, stream>>>(x, agg, deg, Wl0, b0, Wr0,
                                                      h1, 96, 96, 6, 1);
  // layer 1: h1(96) -> h2(96), relu
  zero_f32<<<(aggN + B - 1) / B, B, 0, stream>>>(agg, aggN);
  scatter_kernel<<<(int)((scatN + B - 1) / B), B, 0, stream>>>(h1, src, dst, agg);
  sage_gemm<<<(MTILES * 6 + 3) / 4, 128, 0, stream>>>(h1, agg, deg, Wl1, b1, Wr1,
                                                      h2, 96, 96, 6, 1);
  // layer 2: h2(96) -> h3(64), no relu
  zero_f32<<<(aggN + B - 1) / B, B, 0, stream>>>(agg, aggN);
  scatter_kernel<<<(int)((scatN + B - 1) / B), B, 0, stream>>>(h2, src, dst, agg);
  sage_gemm<<<(MTILES * 4 + 3) / 4, 128, 0, stream>>>(h2, agg, deg, Wl2, b2, Wr2,
                                                      h3, 96, 64, 4, 0);

  // global mean pool
  pool_kernel<<<(int)((poolN + B - 1) / B), B, 0, stream>>>(h3, batch, out, cnt);
  pool_div<<<(NUM_GRAPHS * 64 + B - 1) / B, B, 0, stream>>>(out, cnt);
}